// NodeEncoder_with_interpolation_7052336300122
// MI455X (gfx1250) — compile-verified
//
#include <hip/hip_runtime.h>
#include <hip/hip_bf16.h>
#include <stdint.h>
#include <stddef.h>

typedef float v4f __attribute__((ext_vector_type(4)));
typedef int   v4i __attribute__((ext_vector_type(4)));
// address-space-qualified int4 for the async-copy builtin:
// param 1 is AS1 (global, printed as `__device__` by clang), param 2 is AS3 (LDS)
typedef __attribute__((address_space(1))) v4i g_v4i;
typedef __attribute__((address_space(3))) v4i l_v4i;

// ZS table from the reference (sorted, 16 entries)
#define ZS_INIT {0.f,1.f,6.f,7.f,8.f,11.f,13.f,14.f,16.f,17.f,26.f,29.f,47.f,78.f,79.f,83.f}
// 1 / (ZS[j] - ZS[j-1]) precomputed at compile time (j=0 unused -> 1)
#define ZINV_INIT {1.f, 1.f/1.f, 1.f/5.f, 1.f/1.f, 1.f/1.f, 1.f/3.f, 1.f/2.f, 1.f/1.f, \
                   1.f/2.f, 1.f/1.f, 1.f/9.f, 1.f/3.f, 1.f/18.f, 1.f/31.f, 1.f/1.f, 1.f/4.f}

__global__ __launch_bounds__(256)
void node_encoder_kernel(const float* __restrict__ z, float* __restrict__ out,
                         int n, int ntiles)
{
    constexpr float czs[16]  = ZS_INIT;   // literal constants for the compare chain
    constexpr float cinv[16] = ZINV_INIT;

    // fused (zhi, zlo, 1/denom, 0) per index: one ds_load_b128 per element,
    // 16 entries x 16 B span exactly the 64 LDS banks -> conflict-free.
    __shared__ v4f tab4[16];
    __shared__ v4f stage[2][256];         // double-buffered async staging (8 KB)

    const int tid = threadIdx.x;
    if (tid < 16) {
        int jm = (tid > 0) ? tid - 1 : 0;
        v4f tv = { czs[tid], czs[jm], cinv[tid], 0.f };
        tab4[tid] = tv;
    }
    __syncthreads();

    const int stride = gridDim.x;

    // Issue one 4 KB tile (1024 floats) as a per-lane async b128 copy into stage[buf].
    auto issue_tile = [&](int t, int buf) {
        if (t >= ntiles) t = ntiles - 1;  // harmless re-read of last tile
        int base = t * 1024 + tid * 4;
        if (base > n - 4) base = n - 4;   // clamp tail lanes in-bounds (n % 4 == 0)
        const float* gp = z + base;
        v4f* lp = &stage[buf][tid];
        __builtin_amdgcn_global_load_async_to_lds_b128(
            (g_v4i*)(v4i*)(void*)const_cast<float*>(gp),
            (l_v4i*)(v4i*)(void*)lp,
            /*imm offset*/0, /*cpol*/0);
    };

    // Encode one element: searchsorted (branchless literal compares), fused LDS
    // lookup, division-free weights, cndmask row build, 4x NT b128 stores.
    auto encode = [&](float zz, float* orow) {
        int j = 0;
        #pragma unroll
        for (int k = 0; k < 16; ++k) j += (czs[k] < zz) ? 1 : 0;
        j = (j > 15) ? 15 : j;

        v4f t = tab4[j];
        float zhi = t.x, zlo = t.y, invd = t.z;
        float whi = (zhi == zz) ? 1.0f : (zz - zlo) * invd;
        float wlo = (zhi - zz) * invd;    // exactly 0 on exact match

        float r[16];
        #pragma unroll
        for (int k = 0; k < 16; ++k) {
            float v = (k == j - 1) ? wlo : 0.0f;  // w_lo at j-1 first...
            r[k] = (k == j) ? whi : v;            // ...then w_hi at j wins
        }
        #pragma unroll
        for (int q = 0; q < 4; ++q) {
            v4f vv = { r[4*q+0], r[4*q+1], r[4*q+2], r[4*q+3] };
            __builtin_nontemporal_store(vv, (v4f*)(orow + 4*q));
        }
    };

    int tile = blockIdx.x;
    int p = 0;
    issue_tile(tile, p);                  // prologue: 1 copy in flight

    for (; tile < ntiles; tile += stride) {
        issue_tile(tile + stride, p ^ 1); // keep pipeline 2 deep
        // Async loads complete in order: <=1 outstanding means the current
        // tile's LDS data has landed (only the prefetched one remains).
        asm volatile("s_wait_asynccnt 0x1" ::: "memory");

        v4f zv = stage[p][tid];           // ds_load_b128, lane-private bytes
        int base = tile * 1024 + tid * 4;

        if (tile * 1024 + 1024 <= n) {    // block-uniform full-tile fast path
            #pragma unroll
            for (int s = 0; s < 4; ++s)
                encode(zv[s], out + (size_t)(base + s) * 16);
        } else {                          // guarded tail path
            #pragma unroll
            for (int s = 0; s < 4; ++s)
                if (base + s < n)
                    encode(zv[s], out + (size_t)(base + s) * 16);
        }
        p ^= 1;
    }
}

extern "C" void kernel_launch(void* const* d_in, const int* in_sizes, int n_in,
                              void* d_out, int out_size, void* d_ws, size_t ws_size,
                              hipStream_t stream) {
    const float* z = (const float*)d_in[0];
    float* out = (float*)d_out;
    int n = in_sizes[0];

    int ntiles = (n + 1023) / 1024;       // 1024 elements (4 KB) per tile
    if (ntiles < 1) ntiles = 1;
    int grid = ntiles < 4096 ? ntiles : 4096;  // persistent blocks, grid-stride tiles

    node_encoder_kernel<<<grid, 256, 0, stream>>>(z, out, n, ntiles);
}